// EquivMPBlock_51762945851635
// MI455X (gfx1250) — compile-verified
//
#include <hip/hip_runtime.h>
#include <hip/hip_bf16.h>

// ---------------------------------------------------------------------------
// EquivMPBlock on gfx1250 (MI455X), fp32 end-to-end.
// Edge pass: 5 batched GEMMs via V_WMMA_F32_16X16X4_F32 (one 16-edge tile per
// wave32), scatter-add via global_atomic_add_f32 into m_agg (L2-resident).
// Node pass: Ws/Wv GEMMs via WMMA + residual/ReLU.
// Weights staged transposed in LDS so every B-fragment is a ds_load_b64.
// ---------------------------------------------------------------------------

typedef __attribute__((ext_vector_type(2))) float v2f;
typedef __attribute__((ext_vector_type(8))) float v8f;

#define INV3f  0.5773502691896258f   // 1/sqrt(3)
#define INV6f  0.4082482904638631f   // 1/sqrt(6)
#define C0f    0.10206207261596575f  // 1/sqrt(96)
#define C1f    0.15309310892394862f  // sqrt(3/128)
#define LIN_Sf 0.125f                // 1/sqrt(64)
#define LIN_Vf 0.17677669529663687f  // 1/sqrt(32)

static __device__ inline v8f wmma4(v2f a, v2f b, v8f c) {
  // D(16x16,f32) = A(16x4,f32) x B(4x16,f32) + C
  return __builtin_amdgcn_wmma_f32_16x16x4_f32(
      /*neg_a=*/false, a, /*neg_b=*/false, b,
      /*c_mod=*/(short)0, c, /*reuse_a=*/false, /*reuse_b=*/false);
}

// ---------------------------------------------------------------------------
// Edge kernel: 8 waves / block, each wave owns a tile of 16 edges.
// LDS weight layout (all transposed to [n][k] so B-frag j=0,1 are contiguous):
//   W1t @0     (64n x 64k)   W2t @4096 (64n x 32k)   W3t @6144 (32n x 64k)
//   W4t @8192  (32n x 32k)   W5t @9216 (32n x 32k)
// ---------------------------------------------------------------------------
__global__ __launch_bounds__(256) void edge_kernel(
    const float* __restrict__ x, const float* __restrict__ sh,
    const int* __restrict__ ei,
    const float* __restrict__ W1, const float* __restrict__ W2,
    const float* __restrict__ W3, const float* __restrict__ W4,
    const float* __restrict__ W5,
    float* __restrict__ magg, int E_, int ntiles)
{
  __shared__ float w[10240];
  for (int idx = threadIdx.x; idx < 4096; idx += 256) { int k = idx >> 6, n = idx & 63; w[n * 64 + k] = W1[idx]; }
  for (int idx = threadIdx.x; idx < 2048; idx += 256) { int k = idx >> 6, n = idx & 63; w[4096 + n * 32 + k] = W2[idx]; }
  for (int idx = threadIdx.x; idx < 2048; idx += 256) { int k = idx >> 5, n = idx & 31; w[6144 + n * 64 + k] = W3[idx]; }
  for (int idx = threadIdx.x; idx < 1024; idx += 256) { int k = idx >> 5, n = idx & 31; w[8192 + n * 32 + k] = W4[idx]; }
  for (int idx = threadIdx.x; idx < 1024; idx += 256) { int k = idx >> 5, n = idx & 31; w[9216 + n * 32 + k] = W5[idx]; }
  __syncthreads();

  const int lane = threadIdx.x & 31;
  const int m    = lane & 15;   // edge-in-tile (A row / output col)
  const int hi   = lane >> 4;   // K-stripe half (A: K=4k+2hi+{0,1})
  const int tile = blockIdx.x * 8 + (threadIdx.x >> 5);
  if (tile >= ntiles) return;   // wave-uniform; EXEC stays all-ones for WMMA

  const int e    = tile * 16 + m;
  const int srci = ei[e];
  const int dsti = ei[E_ + e];
  const float4 she = *(const float4*)&sh[4 * e];
  const float sh0 = she.x, s1x = she.y, s1y = she.z, s1z = she.w;
  const float* xr = x + (size_t)srci * 160;

  // ---- GEMM1 ((ss*e0)@W1 -> acc0) and GEMM3 (ss@W3 -> acc3), shared loads
  v8f acc0[4] = {};
  v8f acc3[2] = {};
#pragma unroll
  for (int k = 0; k < 16; ++k) {
    const int c = 4 * k + 2 * hi;
    v2f s  = *(const v2f*)&xr[c];      // s[src][c..c+1], L2 hit
    v2f a1 = s * sh0;
#pragma unroll
    for (int n = 0; n < 4; ++n) {
      v2f b = *(const v2f*)&w[(16 * n + m) * 64 + c];
      acc0[n] = wmma4(a1, b, acc0[n]);
    }
#pragma unroll
    for (int n = 0; n < 2; ++n) {
      v2f b = *(const v2f*)&w[6144 + (16 * n + m) * 64 + c];
      acc3[n] = wmma4(s, b, acc3[n]);
    }
  }

  // ---- GEMM2 ((dot*INV3)@W2 -> acc0), dot folded into A-frag build
#pragma unroll
  for (int k = 0; k < 8; ++k) {
    const int u0 = 4 * k + 2 * hi;
    const float* vp = xr + 64 + u0 * 3;
    v2f a;
    a.x = (vp[0] * s1x + vp[1] * s1y + vp[2] * s1z) * INV3f;
    a.y = (vp[3] * s1x + vp[4] * s1y + vp[5] * s1z) * INV3f;
#pragma unroll
    for (int n = 0; n < 4; ++n) {
      v2f b = *(const v2f*)&w[4096 + (16 * n + m) * 32 + u0];
      acc0[n] = wmma4(a, b, acc0[n]);
    }
  }

  // Per-output-row (M = r + 8*hi) broadcasts of dst index and e1 components.
  int dr[8]; float e1M[3][8];
#pragma unroll
  for (int r = 0; r < 8; ++r) {
    const int M = r + 8 * hi;
    dr[r]     = __shfl(dsti, M, 32);
    e1M[0][r] = __shfl(s1x, M, 32);
    e1M[1][r] = __shfl(s1y, M, 32);
    e1M[2][r] = __shfl(s1z, M, 32);
  }

  // ---- scatter out0 = C0 * acc0  (cols 0..63 of message)
#pragma unroll
  for (int n = 0; n < 4; ++n)
#pragma unroll
    for (int r = 0; r < 8; ++r)
      atomicAdd(&magg[(size_t)dr[r] * 160 + 16 * n + m], C0f * acc0[n][r]);

  // ---- per xyz component: GEMM4 + GEMM5 -> acc, combine with p3 (acc3)
  const float sh0i3 = sh0 * INV3f;
  const float s1c[3] = {s1x, s1y, s1z};
#pragma unroll
  for (int i = 0; i < 3; ++i) {
    const int i1 = (i + 1) % 3, i2 = (i + 2) % 3;
    v8f acc[2] = {};
#pragma unroll
    for (int k = 0; k < 8; ++k) {
      const int u0 = 4 * k + 2 * hi;
      const float* vp = xr + 64 + u0 * 3;
      v2f a4, a5;
      a4.x = vp[i]     * sh0i3;                                    // vv*e0*INV3
      a4.y = vp[3 + i] * sh0i3;
      a5.x = (vp[i1]     * s1c[i2] - vp[i2]     * s1c[i1]) * INV6f; // cross*INV6
      a5.y = (vp[3 + i1] * s1c[i2] - vp[3 + i2] * s1c[i1]) * INV6f;
#pragma unroll
      for (int n = 0; n < 2; ++n) {
        v2f b4 = *(const v2f*)&w[8192 + (16 * n + m) * 32 + u0];
        acc[n] = wmma4(a4, b4, acc[n]);
        v2f b5 = *(const v2f*)&w[9216 + (16 * n + m) * 32 + u0];
        acc[n] = wmma4(a5, b5, acc[n]);
      }
    }
#pragma unroll
    for (int n = 0; n < 2; ++n)
#pragma unroll
      for (int r = 0; r < 8; ++r) {
        const float val = C1f * (acc3[n][r] * e1M[i][r] * INV3f + acc[n][r]);
        atomicAdd(&magg[(size_t)dr[r] * 160 + 64 + (16 * n + m) * 3 + i], val);
      }
  }
}

// ---------------------------------------------------------------------------
// Scalar tail kernel for E % 16 edges (0 blocks for E = 800000).
// ---------------------------------------------------------------------------
__global__ void edge_tail_kernel(
    const float* __restrict__ x, const float* __restrict__ sh,
    const int* __restrict__ ei,
    const float* __restrict__ W1, const float* __restrict__ W2,
    const float* __restrict__ W3, const float* __restrict__ W4,
    const float* __restrict__ W5,
    float* __restrict__ magg, int E_, int start)
{
  int e = start + blockIdx.x * blockDim.x + threadIdx.x;
  if (e >= E_) return;
  int srci = ei[e], dsti = ei[E_ + e];
  const float* xr = x + (size_t)srci * 160;
  float sh0 = sh[4 * e];
  float s1[3] = {sh[4 * e + 1], sh[4 * e + 2], sh[4 * e + 3]};
  float dot[32];
  for (int u = 0; u < 32; ++u)
    dot[u] = xr[64 + u * 3] * s1[0] + xr[64 + u * 3 + 1] * s1[1] + xr[64 + u * 3 + 2] * s1[2];
  for (int n = 0; n < 64; ++n) {
    float acc = 0.f;
    for (int k = 0; k < 64; ++k) acc += xr[k] * sh0 * W1[k * 64 + n];
    for (int u = 0; u < 32; ++u) acc += dot[u] * INV3f * W2[u * 64 + n];
    atomicAdd(&magg[(size_t)dsti * 160 + n], C0f * acc);
  }
  for (int wq = 0; wq < 32; ++wq) {
    float sw = 0.f;
    for (int k = 0; k < 64; ++k) sw += xr[k] * W3[k * 32 + wq];
    for (int i = 0; i < 3; ++i) {
      int i1 = (i + 1) % 3, i2 = (i + 2) % 3;
      float acc = sw * s1[i] * INV3f;
      for (int u = 0; u < 32; ++u) {
        float p4 = xr[64 + u * 3 + i] * sh0 * INV3f;
        float p5 = (xr[64 + u * 3 + i1] * s1[i2] - xr[64 + u * 3 + i2] * s1[i1]) * INV6f;
        acc += p4 * W4[u * 32 + wq] + p5 * W5[u * 32 + wq];
      }
      atomicAdd(&magg[(size_t)dsti * 160 + 64 + wq * 3 + i], C1f * acc);
    }
  }
}

// ---------------------------------------------------------------------------
// Node kernel: h = [m_agg_s@Ws*LIN_S | einsum(m_agg_v,Wv)*LIN_V]; out=x+relu(h)
// ---------------------------------------------------------------------------
__global__ __launch_bounds__(256) void node_kernel(
    const float* __restrict__ magg, const float* __restrict__ x,
    const float* __restrict__ Ws, const float* __restrict__ Wv,
    float* __restrict__ out, int N_, int ntiles)
{
  __shared__ float w[5120];  // Wst [64n x 64k] @0, Wvt [32n x 32k] @4096
  for (int idx = threadIdx.x; idx < 4096; idx += 256) { int k = idx >> 6, n = idx & 63; w[n * 64 + k] = Ws[idx]; }
  for (int idx = threadIdx.x; idx < 1024; idx += 256) { int k = idx >> 5, n = idx & 31; w[4096 + n * 32 + k] = Wv[idx]; }
  __syncthreads();

  const int lane = threadIdx.x & 31;
  const int m    = lane & 15;
  const int hi   = lane >> 4;
  const int tile = blockIdx.x * 8 + (threadIdx.x >> 5);
  if (tile >= ntiles) return;
  const int base = tile * 16;
  const int rowm = min(base + m, N_ - 1);  // clamp loads; stores are guarded
  const float* mr = magg + (size_t)rowm * 160;

  // scalar part: (16 x 64) @ Ws
  v8f accs[4] = {};
#pragma unroll
  for (int k = 0; k < 16; ++k) {
    const int c = 4 * k + 2 * hi;
    v2f a = *(const v2f*)&mr[c];
#pragma unroll
    for (int n = 0; n < 4; ++n) {
      v2f b = *(const v2f*)&w[(16 * n + m) * 64 + c];
      accs[n] = wmma4(a, b, accs[n]);
    }
  }
#pragma unroll
  for (int n = 0; n < 4; ++n)
#pragma unroll
    for (int r = 0; r < 8; ++r) {
      const int M = base + r + 8 * hi;
      if (M < N_) {
        size_t idx = (size_t)M * 160 + 16 * n + m;
        float h = accs[n][r] * LIN_Sf;
        out[idx] = x[idx] + fmaxf(h, 0.f);
      }
    }

  // vector part: per component i, (16 x 32) @ Wv
#pragma unroll
  for (int i = 0; i < 3; ++i) {
    v8f av[2] = {};
#pragma unroll
    for (int k = 0; k < 8; ++k) {
      const int u0 = 4 * k + 2 * hi;
      v2f a;
      a.x = mr[64 + u0 * 3 + i];
      a.y = mr[64 + (u0 + 1) * 3 + i];
#pragma unroll
      for (int n = 0; n < 2; ++n) {
        v2f b = *(const v2f*)&w[4096 + (16 * n + m) * 32 + u0];
        av[n] = wmma4(a, b, av[n]);
      }
    }
#pragma unroll
    for (int n = 0; n < 2; ++n)
#pragma unroll
      for (int r = 0; r < 8; ++r) {
        const int M = base + r + 8 * hi;
        if (M < N_) {
          size_t idx = (size_t)M * 160 + 64 + (16 * n + m) * 3 + i;
          float h = av[n][r] * LIN_Vf;
          out[idx] = x[idx] + fmaxf(h, 0.f);
        }
      }
  }
}

// ---------------------------------------------------------------------------
extern "C" void kernel_launch(void* const* d_in, const int* in_sizes, int n_in,
                              void* d_out, int out_size, void* d_ws, size_t ws_size,
                              hipStream_t stream) {
  const float* x  = (const float*)d_in[0];
  const float* sh = (const float*)d_in[1];
  const float* W1 = (const float*)d_in[2];
  const float* W2 = (const float*)d_in[3];
  const float* W3 = (const float*)d_in[4];
  const float* W4 = (const float*)d_in[5];
  const float* W5 = (const float*)d_in[6];
  const float* Ws = (const float*)d_in[7];
  const float* Wv = (const float*)d_in[8];
  const int*   ei = (const int*)d_in[9];   // edge_index (2, E) flat
  float* out = (float*)d_out;

  const int N_ = in_sizes[0] / 160;
  const int E_ = in_sizes[9] / 2;

  float* magg = (float*)d_ws;              // (N, 160) fp32 scratch, L2-resident
  hipMemsetAsync(magg, 0, (size_t)N_ * 160 * sizeof(float), stream);

  const int E16    = E_ & ~15;
  const int ntiles = E16 / 16;
  const int eblk   = (ntiles + 7) / 8;
  if (eblk > 0)
    edge_kernel<<<eblk, 256, 0, stream>>>(x, sh, ei, W1, W2, W3, W4, W5, magg, E_, ntiles);

  const int rem = E_ - E16;
  if (rem > 0)
    edge_tail_kernel<<<(rem + 63) / 64, 64, 0, stream>>>(x, sh, ei, W1, W2, W3, W4, W5, magg, E_, E16);

  const int ntN  = (N_ + 15) / 16;
  const int nblk = (ntN + 7) / 8;
  node_kernel<<<nblk, 256, 0, stream>>>(magg, x, Ws, Wv, out, N_, ntN);
}